// MultilabelClassifier_v2_10840497455644
// MI455X (gfx1250) — compile-verified
//
#include <hip/hip_runtime.h>
#include <cmath>

typedef __attribute__((ext_vector_type(2))) float v2f;
typedef __attribute__((ext_vector_type(8))) float v8f;

__device__ __forceinline__ v8f wmma4(v2f a, v2f b, v8f c) {
  // D = A(16x4 f32) * B(4x16 f32) + C(16x16 f32)
  return __builtin_amdgcn_wmma_f32_16x16x4_f32(
      /*neg_a=*/false, a, /*neg_b=*/false, b,
      /*c_mod=*/(short)0, c, /*reuse_a=*/false, /*reuse_b=*/false);
}

// ---------------------------------------------------------------------------
// Stage 0: transpose local_features (B,C,H,W) -> (B,H,W,C) so ROI gathers are
// channel-contiguous (coalesced).  LDS 32x33 tile, conflict-free.
// grid (32 c-tiles, 512 = b*32+y), block (32,8)
// ---------------------------------------------------------------------------
__global__ void transpose_chw_hwc(const float* __restrict__ in, float* __restrict__ out) {
  __shared__ float tile[32][33];
  const int c0 = blockIdx.x * 32;
  const int by = blockIdx.y;
  const int b = by >> 5, y = by & 31;
  const int tx = threadIdx.x, ty = threadIdx.y;
#pragma unroll
  for (int j = 0; j < 4; ++j) {
    const int c = c0 + ty + j * 8;
    tile[ty + j * 8][tx] = in[(((size_t)b * 1024 + c) * 32 + y) * 32 + tx];
  }
  __syncthreads();
#pragma unroll
  for (int j = 0; j < 4; ++j) {
    const int x = ty + j * 8;
    out[(((size_t)b * 32 + y) * 32 + x) * 1024 + c0 + tx] = tile[tx][x];
  }
}

// ---------------------------------------------------------------------------
// Stage 0b: feats buffer initialized with bm broadcast (GEMM-C accumulates
// into it with fp32 atomics).  16*36*512 = 294912 elements, 1152 blocks x 256.
// ---------------------------------------------------------------------------
__global__ void init_feats(const float* __restrict__ bm, float* __restrict__ feats) {
  const int idx = blockIdx.x * 256 + threadIdx.x;
  feats[idx] = bm[idx % (36 * 512)];
}

// ---------------------------------------------------------------------------
// Stage 1: ROI align (torchvision aligned=False semantics, SR=2, P=3).
// One block per (b,n,p); 256 threads cover 1024 channels in 4 chunks.
// Output roi layout [n][b][p][c] so each n's GEMM A-matrix (144x1024) is
// contiguous with K fastest.
// ---------------------------------------------------------------------------
__global__ void roi_align_k(const float* __restrict__ fT, const float* __restrict__ boxes,
                            float* __restrict__ roi) {
  const int blk = blockIdx.x;
  const int p = blk % 9;
  const int n = (blk / 9) % 36;
  const int b = blk / (9 * 36);
  const int py = p / 3, px = p % 3;
  const float* bx = boxes + ((size_t)b * 36 + n) * 4;
  const float x1 = bx[0] * 32.f, y1 = bx[1] * 32.f;
  const float x2 = bx[2] * 32.f, y2 = bx[3] * 32.f;
  const float rw = fmaxf(x2 - x1, 1.f), rh = fmaxf(y2 - y1, 1.f);

  float w00[4], w01[4], w10[4], w11[4];
  int o00[4], o01[4], o10[4], o11[4];
#pragma unroll
  for (int s = 0; s < 4; ++s) {
    const int sy = s >> 1, sx = s & 1;
    const int iy = py * 2 + sy, ix = px * 2 + sx;
    const float yo = y1 + rh * ((iy + 0.5f) / 6.f);
    const float xo = x1 + rw * ((ix + 0.5f) / 6.f);
    const float vld = (yo >= -1.f && yo <= 32.f && xo >= -1.f && xo <= 32.f) ? 1.f : 0.f;
    const float y = fmaxf(yo, 0.f), x = fmaxf(xo, 0.f);
    const int fy = (int)floorf(y), fx = (int)floorf(x);
    const bool ty = fy >= 31, txc = fx >= 31;
    const int yl = ty ? 31 : fy, yh = ty ? 31 : fy + 1;
    const int xl = txc ? 31 : fx, xh = txc ? 31 : fx + 1;
    const float ly = ty ? 0.f : y - (float)fy;
    const float lx = txc ? 0.f : x - (float)fx;
    const float hy = 1.f - ly, hx = 1.f - lx;
    w00[s] = hy * hx * vld; w01[s] = hy * lx * vld;
    w10[s] = ly * hx * vld; w11[s] = ly * lx * vld;
    o00[s] = (yl * 32 + xl) * 1024; o01[s] = (yl * 32 + xh) * 1024;
    o10[s] = (yh * 32 + xl) * 1024; o11[s] = (yh * 32 + xh) * 1024;
  }
  const float* fb = fT + (size_t)b * 32 * 32 * 1024;
  float* rp = roi + (((size_t)n * 16 + b) * 9 + p) * 1024;
  for (int cc = 0; cc < 4; ++cc) {
    const int c = threadIdx.x + cc * 256;
    float acc = 0.f;
#pragma unroll
    for (int s = 0; s < 4; ++s) {
      acc += w00[s] * fb[o00[s] + c] + w01[s] * fb[o01[s] + c]
           + w10[s] * fb[o10[s] + c] + w11[s] * fb[o11[s] + c];
    }
    rp[c] = acc * 0.25f;
  }
}

// ---------------------------------------------------------------------------
// Stage 2: x = relu(concat([gf.Wg[n]+bg, roi.Wl[n]+bl])), stored [n][b][5120].
// One wave = one 16(M) x 64(N) tile via 4x V_WMMA_F32_16X16X4_F32 per K-step.
// M-tile 0 -> global path (gf, Wg, bg); tiles 1..9 -> roi rows (144x1024).
// 2880 waves total = 360 blocks x 8 waves.
// ---------------------------------------------------------------------------
__global__ void gemm_x(const float* __restrict__ gf, const float* __restrict__ roi,
                       const float* __restrict__ Wg, const float* __restrict__ bg,
                       const float* __restrict__ Wl, const float* __restrict__ bl,
                       float* __restrict__ xbuf) {
  const int wave = blockIdx.x * 8 + (threadIdx.x >> 5);
  const int lane = threadIdx.x & 31;
  const int n = wave / 80;
  const int rem = wave % 80;
  const int mt = rem >> 3;              // 0..9 (M tile)
  const int colb = (rem & 7) * 64;      // N group of 64
  const int grp = lane >> 4, r = lane & 15;

  const float* A; const float* Bw; const float* bias; int row_base;
  if (mt == 0) { A = gf;  row_base = 0;            Bw = Wg + (size_t)n * 1024 * 512; bias = bg + n * 512; }
  else         { A = roi + (size_t)n * 144 * 1024; row_base = (mt - 1) * 16;
                 Bw = Wl + (size_t)n * 1024 * 512; bias = bl + n * 512; }

  const float* arow = A + (size_t)(row_base + r) * 1024 + 2 * grp;
  const float* bcol = Bw + (size_t)(2 * grp) * 512 + colb + r;

  v8f acc0 = {}, acc1 = {}, acc2 = {}, acc3 = {};
  for (int k0 = 0; k0 < 1024; k0 += 4) {
    v2f a; a.x = arow[k0]; a.y = arow[k0 + 1];
    const float* bp = bcol + (size_t)k0 * 512;
    __builtin_prefetch(bp + 16 * 512, 0, 1);
    v2f b0; b0.x = bp[0];  b0.y = bp[512];
    v2f b1; b1.x = bp[16]; b1.y = bp[528];
    v2f b2; b2.x = bp[32]; b2.y = bp[544];
    v2f b3; b3.x = bp[48]; b3.y = bp[560];
    acc0 = wmma4(a, b0, acc0);
    acc1 = wmma4(a, b1, acc1);
    acc2 = wmma4(a, b2, acc2);
    acc3 = wmma4(a, b3, acc3);
  }
#pragma unroll
  for (int t = 0; t < 4; ++t) {
    v8f acc = t == 0 ? acc0 : t == 1 ? acc1 : t == 2 ? acc2 : acc3;
    const int col = colb + t * 16 + r;
    const float bv = bias[col];
#pragma unroll
    for (int v = 0; v < 8; ++v) {
      const int rl = v + 8 * grp;       // local row in tile
      float val = acc[v] + bv;
      val = val > 0.f ? val : 0.f;      // relu
      size_t idx;
      if (mt == 0) {
        idx = ((size_t)n * 16 + rl) * 5120 + col;          // p = 0 slot
      } else {
        const int rr = row_base + rl;                      // b*9 + (p-1)
        const int b = rr / 9;
        const int p = rr - b * 9 + 1;
        idx = ((size_t)n * 16 + b) * 5120 + (size_t)p * 512 + col;
      }
      xbuf[idx] = val;
    }
  }
}

// ---------------------------------------------------------------------------
// Stage 3: feats[b][n][h] += x[n](16x5120) . Wm[n](5120x512).
// K split into 8 chunks of 640 so enough waves stream the 377 MB of Wm to
// saturate HBM; partials land via global_atomic_add_f32 on the bm-initialized
// feats buffer.  2304 waves = 288 blocks x 8 waves.
// ---------------------------------------------------------------------------
__global__ void gemm_feats(const float* __restrict__ xbuf, const float* __restrict__ Wm,
                           float* __restrict__ feats) {
  const int wave = blockIdx.x * 8 + (threadIdx.x >> 5);
  const int lane = threadIdx.x & 31;
  const int n = wave / 64;
  const int rem = wave % 64;
  const int colb = (rem >> 3) * 64;
  const int kbase = (rem & 7) * 640;
  const int grp = lane >> 4, r = lane & 15;

  const float* arow = xbuf + ((size_t)n * 16 + r) * 5120 + 2 * grp;
  const float* bcol = Wm + (size_t)n * 5120 * 512 + (size_t)(2 * grp) * 512 + colb + r;

  v8f acc0 = {}, acc1 = {}, acc2 = {}, acc3 = {};
  for (int k0 = kbase; k0 < kbase + 640; k0 += 4) {
    v2f a; a.x = arow[k0]; a.y = arow[k0 + 1];
    const float* bp = bcol + (size_t)k0 * 512;
    __builtin_prefetch(bp + 16 * 512, 0, 1);
    v2f b0; b0.x = bp[0];  b0.y = bp[512];
    v2f b1; b1.x = bp[16]; b1.y = bp[528];
    v2f b2; b2.x = bp[32]; b2.y = bp[544];
    v2f b3; b3.x = bp[48]; b3.y = bp[560];
    acc0 = wmma4(a, b0, acc0);
    acc1 = wmma4(a, b1, acc1);
    acc2 = wmma4(a, b2, acc2);
    acc3 = wmma4(a, b3, acc3);
  }
#pragma unroll
  for (int t = 0; t < 4; ++t) {
    v8f acc = t == 0 ? acc0 : t == 1 ? acc1 : t == 2 ? acc2 : acc3;
    const int col = colb + t * 16 + r;
#pragma unroll
    for (int v = 0; v < 8; ++v) {
      const int bb = v + 8 * grp;       // batch row
      unsafeAtomicAdd(&feats[((size_t)bb * 36 + n) * 512 + col], acc[v]);
    }
  }
}

// ---------------------------------------------------------------------------
// Stage 4: heads.  loc (B,36,5) then grp (B,12,10), concatenated -> (B,300).
// Tiny (~9 MFLOP): one thread per output element.
// ---------------------------------------------------------------------------
__global__ void heads(const float* __restrict__ feats, const float* __restrict__ Wloc,
                      const float* __restrict__ bloc, const float* __restrict__ Wgrp,
                      const float* __restrict__ bgrp, float* __restrict__ out) {
  const int idx = blockIdx.x * 256 + threadIdx.x;
  if (idx >= 16 * 300) return;
  const int b = idx / 300, i = idx % 300;
  float s;
  if (i < 180) {
    const int n = i / 5, l = i % 5;
    const float* f = feats + ((size_t)b * 36 + n) * 512;
    const float* w = Wloc + (size_t)n * 512 * 5 + l;
    s = bloc[n * 5 + l];
    for (int h = 0; h < 512; ++h) s += f[h] * w[h * 5];
  } else {
    const int q = i - 180, g = q / 10, l = q % 10;
    s = bgrp[g * 10 + l];
    for (int j = 0; j < 3; ++j) {
      const float* f = feats + ((size_t)b * 36 + 3 * g + j) * 512;
      const float* w = Wgrp + ((size_t)g * 1536 + j * 512) * 10 + l;
      for (int h = 0; h < 512; ++h) s += f[h] * w[h * 10];
    }
  }
  out[idx] = s;
}

extern "C" void kernel_launch(void* const* d_in, const int* in_sizes, int n_in,
                              void* d_out, int out_size, void* d_ws, size_t ws_size,
                              hipStream_t stream) {
  (void)in_sizes; (void)n_in; (void)out_size; (void)ws_size;
  const float* lf   = (const float*)d_in[0];   // (16,1024,32,32)
  const float* gf   = (const float*)d_in[1];   // (16,1024)
  const float* bbox = (const float*)d_in[2];   // (16,36,4)
  const float* Wg   = (const float*)d_in[3];   // (36,1024,512)
  const float* bg   = (const float*)d_in[4];   // (36,512)
  const float* Wl   = (const float*)d_in[5];   // (36,1024,512)
  const float* bl   = (const float*)d_in[6];   // (36,512)
  const float* Wm   = (const float*)d_in[7];   // (36,5120,512)
  const float* bm   = (const float*)d_in[8];   // (36,512)
  const float* Wloc = (const float*)d_in[9];   // (36,512,5)
  const float* bloc = (const float*)d_in[10];  // (36,5)
  const float* Wgrp = (const float*)d_in[11];  // (12,1536,10)
  const float* bgrp = (const float*)d_in[12];  // (12,10)
  float* out = (float*)d_out;

  char* ws = (char*)d_ws;
  float* fT    = (float*)(ws);                                  // 64 MB  (B,H,W,C)
  float* roi   = (float*)(ws + (size_t)64 * 1024 * 1024);       // 21.25 MB [n][b][p][c]
  float* xbuf  = (float*)(ws + (size_t)86 * 1024 * 1024);       // 11.8 MB [n][b][5120]
  float* feats = (float*)(ws + (size_t)98 * 1024 * 1024);       // 1.18 MB [b][n][512]

  transpose_chw_hwc<<<dim3(32, 512), dim3(32, 8), 0, stream>>>(lf, fT);
  init_feats<<<1152, 256, 0, stream>>>(bm, feats);
  roi_align_k<<<16 * 36 * 9, 256, 0, stream>>>(fT, bbox, roi);
  gemm_x<<<360, 256, 0, stream>>>(gf, roi, Wg, bg, Wl, bl, xbuf);
  gemm_feats<<<288, 256, 0, stream>>>(xbuf, Wm, feats);
  heads<<<19, 256, 0, stream>>>(feats, Wloc, bloc, Wgrp, bgrp, out);
}